// S4D_42734924595310
// MI455X (gfx1250) — compile-verified
//
#include <hip/hip_runtime.h>

typedef __attribute__((ext_vector_type(2))) float v2f;
typedef __attribute__((ext_vector_type(8))) float v8f;
typedef unsigned int uint32x4 __attribute__((ext_vector_type(4)));
typedef int int32x8 __attribute__((ext_vector_type(8)));
typedef int int32x4 __attribute__((ext_vector_type(4)));

#define B_     8
#define L_     4096
#define M_     1024
#define N_     64
#define Q_     64
#define NPAIR_ 32    // two 64-chunks per loop iteration
#define RL     68    // LDS row pitch (words): TDM pads 4 DWORDs per 64 -> conflict-free b64

// LDS byte offset of a __shared__ object (strip the generic aperture).
#define LDS_OFF(p) ((unsigned int)(unsigned long long)(__attribute__((address_space(3))) const void*)(p))

// One TDM descriptor: 4-D tile  x(1 elem) * y(64 rows, stride M) * z(8 batches,
// stride L*M) * zz(2 chunks, stride Q*M), data_size=4B, LDS padding 4 DWORDs
// after every 64 -> lands exactly in Ut[16][RL] (col = b + 8*ch, pitch 68).
__device__ __forceinline__ void tdm_load_pair(const float* u, int m, int t, unsigned int lds_byte)
{
    const unsigned long long ga =
        (unsigned long long)(size_t)u + 4ull * ((unsigned long long)t * (2u * Q_ * M_) + (unsigned int)m);

    uint32x4 g0;
    g0[0] = 1u;                                              // count=1, user descriptor
    g0[1] = lds_byte;                                        // lds_addr (bytes)
    g0[2] = (unsigned int)(ga & 0xFFFFFFFFull);              // global_addr[31:0]
    g0[3] = (unsigned int)((ga >> 32) & 0x1FFFFFFull)        // global_addr[56:32]
          | (2u << 30);                                      // type = 2 ("image")

    int32x8 g1;
    g1[0] = (int)((2u << 16)                                 // data_size = 4B
                | (1u << 20)                                 // pad_enable
                | (5u << 22)                                 // pad_interval: 64 DWORDs
                | (3u << 25));                               // pad_amount:   4 DWORDs
    g1[1] = (int)(1u << 16);                                 // tensor_dim0 = 1
    g1[2] = (int)((unsigned)Q_ << 16);                       // tensor_dim1 = 64
    g1[3] = (int)(1u << 16);                                 // tile_dim0 = 1
    g1[4] = (int)((unsigned)Q_ | ((unsigned)B_ << 16));      // tile_dim1 = 64, tile_dim2 = 8
    g1[5] = (int)M_;                                         // tensor_dim0_stride = M (elems)
    g1[6] = 0;                                               // dim0_stride hi, dim1_stride[15:0]
    g1[7] = (int)(((unsigned)L_ * M_) >> 16);                // tensor_dim1_stride = L*M (elems)

    int32x4 g2;
    g2[0] = B_;                                              // tensor_dim2 = 8
    g2[1] = 2;                                               // tensor_dim3 = 2
    g2[2] = Q_ * M_;                                         // tensor_dim2_stride = Q*M (elems)
    g2[3] = (int)(2u << 16);                                 // tile_dim3 = 2

    int32x4 g3;
    g3[0] = 0; g3[1] = 0; g3[2] = 0; g3[3] = 0;              // no 5th dim

    int32x8 g4;                                              // VADDR4 slot: unused
    g4[0] = 0; g4[1] = 0; g4[2] = 0; g4[3] = 0;
    g4[4] = 0; g4[5] = 0; g4[6] = 0; g4[7] = 0;

    __builtin_amdgcn_tensor_load_to_lds(g0, g1, g2, g3, g4, 0);
}

// One workgroup per channel m. 128 threads = 4 wave32; wave w owns rows
// [16w,16w+16). Chunked diagonal-SSM scan; cols 0-7 = batches of chunk 2t,
// cols 8-15 = chunk 2t+1. TDM double-buffers the U tiles.
__launch_bounds__(128, 4)
__global__ void s4d_chunked_wmma(const float* __restrict__ u,
                                 const float* __restrict__ A_log,
                                 const float* __restrict__ D,
                                 const float* __restrict__ C_re,
                                 const float* __restrict__ log_dt,
                                 float* __restrict__ y)
{
    __shared__ float P[Q_ + 1][N_];          // lam_n^i, i = 0..64
    __shared__ float Ks2[2 * Q_];            // Ks2[64+d] = K[d]; lower half 0 (branchless Toeplitz)
    __shared__ float Csh[N_];
    __shared__ float lamQ[N_];
    __shared__ __align__(16) float Ut[2][16][RL];  // TDM double buffer, Ut[buf][col][j]
    __shared__ float St[16][RL];             // St[col][n]: cols 0-7 = S_old, 8-15 = S_mid

    const int m    = blockIdx.x;
    const int tid  = threadIdx.x;
    const int lane = tid & 31;
    const int w    = tid >> 5;
    const int row0 = w << 4;
    const int col  = lane & 15;              // B/C/D column
    const int hi   = lane >> 4;
    const int ar   = row0 + (lane & 15);     // A-matrix row for this lane

    // kick off DMA of the first U tile pair before doing anything else
    if (w == 0) tdm_load_pair(u, m, 0, LDS_OFF(&Ut[0][0][0]));

    const float dt   = __expf(log_dt[m]);
    const float Dval = D[m];

    // ---- per-channel tables (overlapped with the TDM transfer) ----
    if (tid < N_) {
        const float An = -__expf(A_log[tid]);
        lamQ[tid] = __expf(dt * An * (float)Q_);
        Csh[tid]  = C_re[m * N_ + tid];
    }
    for (int idx = tid; idx < (Q_ + 1) * N_; idx += 128) {
        const int i = idx >> 6, n = idx & 63;
        P[i][n] = __expf(dt * (-__expf(A_log[n])) * (float)i);
    }
    for (int idx = tid; idx < 16 * RL; idx += 128)
        St[idx / RL][idx % RL] = 0.0f;
    __syncthreads();
    if (tid < Q_) {
        float s = 0.0f;
        #pragma unroll
        for (int n = 0; n < N_; ++n) s += Csh[n] * P[tid][n];
        Ks2[Q_ + tid] = s;
        Ks2[tid]      = 0.0f;
    }
    __syncthreads();

    // ---- hoist loop-invariant A-operand fragments into registers ----
    v2f aT[16], aV[16], aW[16];
    #pragma unroll
    for (int kb = 0; kb < 16; ++kb) {
        const int k0 = (kb << 2) + (hi << 1);
        aT[kb].x = Ks2[Q_     + ar - k0];            // T[i][j] = K[i-j] (causal)
        aT[kb].y = Ks2[Q_ - 1 + ar - k0];
        aV[kb].x = Csh[k0]     * P[ar + 1][k0];      // V[i][n] = C_n lam^{i+1}
        aV[kb].y = Csh[k0 + 1] * P[ar + 1][k0 + 1];
        aW[kb].x = P[Q_ - 1 - k0][ar];               // W[n][j] = lam^{Q-1-j}
        aW[kb].y = P[Q_ - 2 - k0][ar];
    }

    const int b_  = col & 7;
    const int ch_ = col >> 3;

    for (int t = 0; t < NPAIR_; ++t) {
        // pipeline: issue DMA for pair t+1, then wait for pair t's DMA.
        // Safe: buf[(t+1)&1]'s previous readers all finished before barrier (4)
        // of iteration t-1, which wave 0 has already passed.
        if (w == 0) {
            if (t + 1 < NPAIR_) {
                tdm_load_pair(u, m, t + 1, LDS_OFF(&Ut[(t + 1) & 1][0][0]));
                __builtin_amdgcn_s_wait_tensorcnt(1);   // in-order: pair t complete
            } else {
                __builtin_amdgcn_s_wait_tensorcnt(0);
            }
        }
        __syncthreads();                                   // (A) publish Ut[t&1] + close prev iter
        const float (*U)[RL] = Ut[t & 1];

        // s = W @ U for both chunks (rows = state n)
        v8f sacc;
        #pragma unroll
        for (int r = 0; r < 8; ++r) sacc[r] = 0.0f;
        #pragma unroll
        for (int kb = 0; kb < 16; ++kb) {
            const int k0 = (kb << 2) + (hi << 1);
            v2f bU; bU.x = U[col][k0]; bU.y = U[col][k0 + 1];
            sacc = __builtin_amdgcn_wmma_f32_16x16x4_f32(false, aW[kb], false, bU,
                                                         (short)0, sacc, false, false);
        }

        // publish S_mid = lam^Q * S_old + s_even into cols 8..15
        if (col < 8) {
            #pragma unroll
            for (int r = 0; r < 8; ++r) {
                const int n = row0 + r + (hi << 3);
                St[col + 8][n] = lamQ[n] * St[col][n] + sacc[r];
            }
        }
        __syncthreads();                                   // (3)

        // y = D*u + T@U + V@S   (two independent WMMA chains for co-execution)
        v8f accT, accV;
        #pragma unroll
        for (int r = 0; r < 8; ++r) {
            accT[r] = Dval * U[col][row0 + r + (hi << 3)];
            accV[r] = 0.0f;
        }
        #pragma unroll
        for (int kb = 0; kb < 16; ++kb) {
            const int k0 = (kb << 2) + (hi << 1);
            v2f bU, bS;
            bU.x = U[col][k0];  bU.y = U[col][k0 + 1];
            bS.x = St[col][k0]; bS.y = St[col][k0 + 1];
            accT = __builtin_amdgcn_wmma_f32_16x16x4_f32(false, aT[kb], false, bU,
                                                         (short)0, accT, false, false);
            accV = __builtin_amdgcn_wmma_f32_16x16x4_f32(false, aV[kb], false, bS,
                                                         (short)0, accV, false, false);
        }
        __syncthreads();                                   // (4) all Ut/St reads done

        // advance: S_old <- lam^Q * S_mid + s_odd   (lanes col>=8 hold s_odd)
        if (col >= 8) {
            #pragma unroll
            for (int r = 0; r < 8; ++r) {
                const int n = row0 + r + (hi << 3);
                St[col - 8][n] = lamQ[n] * St[col][n] + sacc[r];
            }
        }
        // store y[b, (2t+ch)*Q + row, m]
        #pragma unroll
        for (int r = 0; r < 8; ++r) {
            const int row = row0 + r + (hi << 3);
            y[((size_t)b_ * L_ + (size_t)(t * 2 + ch_) * Q_ + row) * M_ + m] = accT[r] + accV[r];
        }
    }
}

extern "C" void kernel_launch(void* const* d_in, const int* in_sizes, int n_in,
                              void* d_out, int out_size, void* d_ws, size_t ws_size,
                              hipStream_t stream) {
    (void)in_sizes; (void)n_in; (void)d_ws; (void)ws_size; (void)out_size;
    const float* u      = (const float*)d_in[0];
    const float* A_log  = (const float*)d_in[1];
    const float* D      = (const float*)d_in[2];
    const float* C_re   = (const float*)d_in[3];
    // d_in[4] = C_im: unused — exp(at) is real, only Re(C) contributes.
    const float* log_dt = (const float*)d_in[5];
    float* y = (float*)d_out;

    hipLaunchKernelGGL(s4d_chunked_wmma, dim3(M_), dim3(128), 0, stream,
                       u, A_log, D, C_re, log_dt, y);
}